// CoPLGCF_74028056314003
// MI455X (gfx1250) — compile-verified
//
#include <hip/hip_runtime.h>
#include <stdint.h>

#define EMB 64
#define TILE 256   // edges staged per block (256 threads = 8 wave32)

// ---------------------------------------------------------------------------
// CDNA5 async global->LDS staging (inline asm: portable across ROCm7.2 /
// amdgpu-toolchain, bypasses the arity-divergent clang builtins).
// GVS mode: VDST holds the LDS address, SADDR base + VADDR offset the global one. Tracked by ASYNCcnt.
// ---------------------------------------------------------------------------
__device__ __forceinline__ void async_g2l_b32(uint32_t lds_byte_off,
                                              uint32_t g_byte_off,
                                              const void* base) {
  asm volatile("global_load_async_to_lds_b32 %0, %1, %2"
               :
               : "v"(lds_byte_off), "v"(g_byte_off),
                 "s"((uint64_t)(uintptr_t)base)
               : "memory");
}

__device__ __forceinline__ void wait_asynccnt0() {
  asm volatile("s_wait_asynccnt 0" ::: "memory");
}

__device__ __forceinline__ int rfl_i32(int x) {
  return __builtin_amdgcn_readfirstlane(x);
}
__device__ __forceinline__ float rfl_f32(float x) {
  return __int_as_float(__builtin_amdgcn_readfirstlane(__float_as_int(x)));
}

// ---------------------------------------------------------------------------
// spmm: dst[row] += val * src[col] per edge. One wave32 per edge: 32 lanes x
// float2 = the 64-wide row. Edge triples staged to LDS with async copies;
// metadata readfirstlane'd to SGPRs so per-edge address math is SALU and the
// gather/scatter use saddr+voffset forms. Gathers hit the L2-resident
// (76.8MB < 192MB L2) src table; scatters are non-returning hardware f32
// atomics absorbed by the L2 atomic units (STOREcnt, no stall).
// ---------------------------------------------------------------------------
__global__ void __launch_bounds__(TILE)
lgcf_spmm(const int* __restrict__ erow, const int* __restrict__ ecol,
          const float* __restrict__ eval, const float* __restrict__ src,
          float* __restrict__ dst, int E) {
  __shared__ int   sh_row[TILE];
  __shared__ int   sh_col[TILE];
  __shared__ float sh_val[TILE];

  const int t        = threadIdx.x;
  const int tileBase = blockIdx.x * TILE;
  const int g        = tileBase + t;

  if (g < E) {
    const uint32_t gb = (uint32_t)g << 2;  // byte offset into edge arrays
    async_g2l_b32((uint32_t)(uintptr_t)&sh_row[t], gb, erow);
    async_g2l_b32((uint32_t)(uintptr_t)&sh_col[t], gb, ecol);
    async_g2l_b32((uint32_t)(uintptr_t)&sh_val[t], gb, eval);
  }
  wait_asynccnt0();
  __syncthreads();

  const int wave   = t >> 5;
  const int lane   = t & 31;
  const int laneo  = lane * 2;           // lane's float offset within the row
  const int nEdge  = (E - tileBase < TILE) ? (E - tileBase) : TILE;
  const int kBeg   = wave * 32;
  int kEnd = kBeg + 32;
  if (kEnd > nEdge) kEnd = nEdge;

#pragma unroll 4
  for (int k = kBeg; k < kEnd; ++k) {
    // wave-uniform LDS broadcast reads -> pin to SGPRs
    const int   col = rfl_i32(sh_col[k]);
    const int   row = rfl_i32(sh_row[k]);
    const float v   = rfl_f32(sh_val[k]);
    const float2 x  = *(const float2*)(src + (size_t)((unsigned)col * EMB) + laneo);
    float* d = dst + (size_t)((unsigned)row * EMB) + laneo;
    unsafeAtomicAdd(d,     x.x * v);   // global_atomic_add_f32 (no return)
    unsafeAtomicAdd(d + 1, x.y * v);
  }
}

// ---------------------------------------------------------------------------
// init: A = concat(user,item); out = concat; B = 0   (float4 streams)
// ---------------------------------------------------------------------------
__global__ void lgcf_init(const float* __restrict__ ue,
                          const float* __restrict__ ie,
                          float* __restrict__ A, float* __restrict__ out,
                          float* __restrict__ B, int totalVec, int userVec) {
  int i = blockIdx.x * blockDim.x + threadIdx.x;
  if (i >= totalVec) return;
  float4 v = (i < userVec) ? ((const float4*)ue)[i]
                           : ((const float4*)ie)[i - userVec];
  ((float4*)A)[i]   = v;
  ((float4*)out)[i] = v;
  ((float4*)B)[i]   = make_float4(0.f, 0.f, 0.f, 0.f);
}

// acc_zero: out += cur; next = 0    (float4 streams)
__global__ void lgcf_acc_zero(float* __restrict__ out,
                              const float* __restrict__ cur,
                              float* __restrict__ nxt, int totalVec) {
  int i = blockIdx.x * blockDim.x + threadIdx.x;
  if (i >= totalVec) return;
  float4 o = ((float4*)out)[i];
  float4 c = ((const float4*)cur)[i];
  o.x += c.x; o.y += c.y; o.z += c.z; o.w += c.w;
  ((float4*)out)[i] = o;
  ((float4*)nxt)[i] = make_float4(0.f, 0.f, 0.f, 0.f);
}

// final: out = (out + cur) * 0.25
__global__ void lgcf_final(float* __restrict__ out,
                           const float* __restrict__ cur, int totalVec) {
  int i = blockIdx.x * blockDim.x + threadIdx.x;
  if (i >= totalVec) return;
  float4 o = ((float4*)out)[i];
  float4 c = ((const float4*)cur)[i];
  o.x = (o.x + c.x) * 0.25f;
  o.y = (o.y + c.y) * 0.25f;
  o.z = (o.z + c.z) * 0.25f;
  o.w = (o.w + c.w) * 0.25f;
  ((float4*)out)[i] = o;
}

extern "C" void kernel_launch(void* const* d_in, const int* in_sizes, int n_in,
                              void* d_out, int out_size, void* d_ws,
                              size_t ws_size, hipStream_t stream) {
  const float* ue   = (const float*)d_in[0];   // (200000, 64) f32
  const float* ie   = (const float*)d_in[1];   // (100000, 64) f32
  const int*   erow = (const int*)d_in[2];     // (9.6M,) i32
  const int*   ecol = (const int*)d_in[3];     // (9.6M,) i32
  const float* eval = (const float*)d_in[4];   // (9.6M,) f32

  const int E    = in_sizes[2];
  const int U    = in_sizes[0] / EMB;
  const int Itm  = in_sizes[1] / EMB;
  const int Ntot = U + Itm;

  float* out = (float*)d_out;            // accumulator, ends as final
  float* A   = (float*)d_ws;             // ego ping buffer (N*64 f32)
  float* B   = A + (size_t)Ntot * EMB;   // ego pong buffer (N*64 f32)

  const int totalVec = Ntot * (EMB / 4); // float4 count
  const int userVec  = U * (EMB / 4);
  const int ewBlocks = (totalVec + 255) / 256;
  const int spBlocks = (E + TILE - 1) / TILE;

  // acc = ego0 ; A = ego0 ; B = 0
  lgcf_init<<<ewBlocks, 256, 0, stream>>>(ue, ie, A, out, B, totalVec, userVec);

  // layer 1: B = S*A ; out += B ; A = 0
  lgcf_spmm<<<spBlocks, TILE, 0, stream>>>(erow, ecol, eval, A, B, E);
  lgcf_acc_zero<<<ewBlocks, 256, 0, stream>>>(out, B, A, totalVec);

  // layer 2: A = S*B ; out += A ; B = 0
  lgcf_spmm<<<spBlocks, TILE, 0, stream>>>(erow, ecol, eval, B, A, E);
  lgcf_acc_zero<<<ewBlocks, 256, 0, stream>>>(out, A, B, totalVec);

  // layer 3: B = S*A ; out = (out + B) / 4
  lgcf_spmm<<<spBlocks, TILE, 0, stream>>>(erow, ecol, eval, A, B, E);
  lgcf_final<<<ewBlocks, 256, 0, stream>>>(out, B, totalVec);
}